// GCN_8881992368460
// MI455X (gfx1250) — compile-verified
//
#include <hip/hip_runtime.h>
#include <hip/hip_bf16.h>

// ---------------------------------------------------------------------------
// GCN forward for MI455X (gfx1250, wave32).
// GEMMs use V_WMMA_F32_16X16X4_F32 (exact fp32), TILES (2 or 1) accumulator chains per
// wave for ILP + A-fragment reuse. SPMM uses global fp32 atomics (dst rows
// are L2-resident: [100000,256] fp32 = 102 MB < 192 MB L2).
// ---------------------------------------------------------------------------

#define N_NODES 100000
#define N_EDGES 1600000
#define NEMB    128
#define NHID    256

typedef __attribute__((ext_vector_type(2))) float v2f;
typedef __attribute__((ext_vector_type(8))) float v8f;

// ---------------------------------------------------------------------------
// Embedding gather: out4[node*32 + c] = table4[feat[node]*32 + c]
// ---------------------------------------------------------------------------
__global__ void gcn_gather_emb(const int* __restrict__ feats,
                               const float4* __restrict__ table4,
                               float4* __restrict__ out4) {
    int idx = blockIdx.x * blockDim.x + threadIdx.x;   // over N_NODES*32 float4
    if (idx >= N_NODES * 32) return;
    int node = idx >> 5;
    int c    = idx & 31;
    out4[idx] = table4[(size_t)feats[node] * 32 + c];
}

// ---------------------------------------------------------------------------
// WMMA fp32 GEMM: C[M,Ncols] = A[M,K] * B[K,Ncols], row-major.
// One block = 8 waves, one 16-row M tile per block. Each wave owns TILES
// adjacent 16-wide n-tiles and accumulates them in the SAME k-loop:
//   - A fragment (LDS) reused across TILES WMMAs
//   - TILES independent accumulator chains hide WMMA latency
// A tile staged in LDS with stride K+1 so per-lane column reads hit distinct
// banks ((K+1) % 64 == 1 for K=128/256). Requires Ncols == 8*TILES*16.
// ---------------------------------------------------------------------------
template <int TILES>
__global__ void gcn_gemm_wmma_f32(const float* __restrict__ A,
                                  const float* __restrict__ B,
                                  float* __restrict__ C,
                                  int K, int Ncols) {
    __shared__ float As[16 * 257];       // max K=256, stride K+1
    const int lds_stride = K + 1;
    const int mBase = blockIdx.x * 16;

    // --- cooperative A-tile load: 16 rows x K floats, coalesced ---
    {
        int r  = threadIdx.x >> 4;               // 0..15
        int c0 = (threadIdx.x & 15) * (K >> 4);  // 8 or 16 floats per thread
        const float* Ap = A + (size_t)(mBase + r) * K + c0;
        float* Sp = &As[r * lds_stride + c0];
        for (int c = 0; c < (K >> 4); c += 4) {
            float4 v = *(const float4*)(Ap + c);
            Sp[c + 0] = v.x; Sp[c + 1] = v.y; Sp[c + 2] = v.z; Sp[c + 3] = v.w;
        }
    }
    __syncthreads();

    const int lane = threadIdx.x & 31;
    const int wave = threadIdx.x >> 5;
    const int col  = lane & 15;                  // N / M sub-index
    const int kOff = (lane < 16) ? 0 : 2;        // K-half split across lane halves
    const int rOff = (lane < 16) ? 0 : 8;        // D-tile row offset
    const int nBase = wave * TILES * 16;         // first n-tile of this wave

    v8f acc[TILES];
#pragma unroll
    for (int t = 0; t < TILES; ++t) acc[t] = (v8f){};

    const float* ap = &As[col * lds_stride + kOff];
    const float* bp = B + (size_t)kOff * Ncols + nBase + col;

#pragma unroll 4
    for (int k0 = 0; k0 < K; k0 += 4) {
        v2f a;
        a.x = ap[0];
        a.y = ap[1];
#pragma unroll
        for (int t = 0; t < TILES; ++t) {
            v2f b;
            b.x = bp[t * 16];
            b.y = bp[t * 16 + Ncols];
            acc[t] = __builtin_amdgcn_wmma_f32_16x16x4_f32(
                false, a, false, b, (short)0, acc[t], false, false);
        }
        ap += 4;
        bp += (size_t)4 * Ncols;
    }

    // D tile store: VGPR r -> row (r + rOff), col
#pragma unroll
    for (int t = 0; t < TILES; ++t) {
        float* Cp = C + (size_t)mBase * Ncols + nBase + t * 16 + col;
#pragma unroll
        for (int r = 0; r < 8; ++r)
            Cp[(size_t)(r + rOff) * Ncols] = acc[t][r];
    }
}

// ---------------------------------------------------------------------------
// SPMM scatter: Y[dst[e], :] += w[e] * X[src[e], :].  One thread = one
// (edge, float4-chunk). Consecutive threads cover one src row -> coalesced.
// shift = log2(D/4).
// ---------------------------------------------------------------------------
__global__ void gcn_spmm_atomic(const int* __restrict__ src,
                                const int* __restrict__ dst,
                                const float* __restrict__ w,
                                const float4* __restrict__ X4,
                                float* __restrict__ Y,
                                int shift, int total) {
    int gid = blockIdx.x * blockDim.x + threadIdx.x;
    if (gid >= total) return;
    int e = gid >> shift;
    int c = gid & ((1 << shift) - 1);
    int s = src[e];
    int d = dst[e];
    float we = w[e];
    float4 x = X4[((size_t)s << shift) + c];
    float* yp = Y + ((((size_t)d << shift) + c) << 2);
    atomicAdd(yp + 0, we * x.x);
    atomicAdd(yp + 1, we * x.y);
    atomicAdd(yp + 2, we * x.z);
    atomicAdd(yp + 3, we * x.w);
}

// ---------------------------------------------------------------------------
// Bias (+optional ReLU), in place over [N_NODES, D], D power of two.
// ---------------------------------------------------------------------------
__global__ void gcn_bias_act(float* __restrict__ Y, const float* __restrict__ b,
                             int mask, int total, int doRelu) {
    int idx = blockIdx.x * blockDim.x + threadIdx.x;
    if (idx >= total) return;
    float v = Y[idx] + b[idx & mask];
    Y[idx] = doRelu ? fmaxf(v, 0.0f) : v;
}

// ---------------------------------------------------------------------------
// Head: out[i] = ((H[i,:] @ Wl1) + bl1) @ Wl2 + bl2.  One thread per node;
// Wl1/Wl2/biases are tiny and L2/L1-hot.
// ---------------------------------------------------------------------------
__global__ void gcn_head(const float* __restrict__ H,
                         const float* __restrict__ Wl1,
                         const float* __restrict__ bl1,
                         const float* __restrict__ Wl2,
                         const float* __restrict__ bl2,
                         float* __restrict__ out) {
    int i = blockIdx.x * blockDim.x + threadIdx.x;
    if (i >= N_NODES) return;
    float acc[16];
#pragma unroll
    for (int j = 0; j < 16; ++j) acc[j] = bl1[j];
    const float4* Hp = (const float4*)(H + (size_t)i * 128);
    for (int k4 = 0; k4 < 32; ++k4) {
        float4 h = Hp[k4];
        const float* wl = Wl1 + k4 * 4 * 16;
#pragma unroll
        for (int j = 0; j < 16; ++j) {
            acc[j] = fmaf(h.x, wl[j +  0], acc[j]);
            acc[j] = fmaf(h.y, wl[j + 16], acc[j]);
            acc[j] = fmaf(h.z, wl[j + 32], acc[j]);
            acc[j] = fmaf(h.w, wl[j + 48], acc[j]);
        }
    }
    float o = bl2[0];
#pragma unroll
    for (int j = 0; j < 16; ++j) o = fmaf(acc[j], Wl2[j], o);
    out[i] = o;
}

// ---------------------------------------------------------------------------
extern "C" void kernel_launch(void* const* d_in, const int* in_sizes, int n_in,
                              void* d_out, int out_size, void* d_ws, size_t ws_size,
                              hipStream_t stream) {
    const int*   features = (const int*)  d_in[0];
    const int*   adj_src  = (const int*)  d_in[1];
    const int*   adj_dst  = (const int*)  d_in[2];
    const float* edge_w   = (const float*)d_in[3];
    const float* emb_tab  = (const float*)d_in[4];
    const float* W1  = (const float*)d_in[5];
    const float* b1  = (const float*)d_in[6];
    const float* W2  = (const float*)d_in[7];
    const float* b2  = (const float*)d_in[8];
    const float* Wl1 = (const float*)d_in[9];
    const float* bl1 = (const float*)d_in[10];
    const float* Wl2 = (const float*)d_in[11];
    const float* bl2 = (const float*)d_in[12];

    float* out_x    = (float*)d_out;             // [N_NODES]
    float* user_emb = out_x + N_NODES;           // [N_NODES,128] output slot, reused as input

    float* buf0 = (float*)d_ws;                  // [N,256] xW1, later [N,128] hW2
    float* buf1 = buf0 + (size_t)N_NODES * NHID; // [N,256] h, later [N,128] h2 accum

    // 1) gather embeddings -> output slot (also GEMM1 input)
    gcn_gather_emb<<<(N_NODES * 32 + 255) / 256, 256, 0, stream>>>(
        features, (const float4*)emb_tab, (float4*)user_emb);

    // 2) GEMM1: xW1 = user_emb @ W1   [N,128]x[128,256]  (8 waves x 2 tiles)
    gcn_gemm_wmma_f32<2><<<N_NODES / 16, 256, 0, stream>>>(user_emb, W1, buf0, NEMB, NHID);

    // 3) SPMM1: h = adj @ xW1  (zero accumulator first)
    hipMemsetAsync(buf1, 0, (size_t)N_NODES * NHID * sizeof(float), stream);
    gcn_spmm_atomic<<<(N_EDGES * (NHID / 4) + 255) / 256, 256, 0, stream>>>(
        adj_src, adj_dst, edge_w, (const float4*)buf0, buf1, 6, N_EDGES * (NHID / 4));

    // 4) + b1, ReLU
    gcn_bias_act<<<(N_NODES * NHID + 255) / 256, 256, 0, stream>>>(
        buf1, b1, NHID - 1, N_NODES * NHID, 1);

    // 5) GEMM2: hW2 = h @ W2   [N,256]x[256,128]  (8 waves x 1 tile) -> buf0
    gcn_gemm_wmma_f32<1><<<N_NODES / 16, 256, 0, stream>>>(buf1, W2, buf0, NHID, 128);

    // 6) SPMM2: h2 = adj @ hW2 -> buf1 (zeroed)
    hipMemsetAsync(buf1, 0, (size_t)N_NODES * 128 * sizeof(float), stream);
    gcn_spmm_atomic<<<(N_EDGES * 32 + 255) / 256, 256, 0, stream>>>(
        adj_src, adj_dst, edge_w, (const float4*)buf0, buf1, 5, N_EDGES * 32);

    // 7) + b2 (no ReLU)
    gcn_bias_act<<<(N_NODES * 128 + 255) / 256, 256, 0, stream>>>(
        buf1, b2, 127, N_NODES * 128, 0);

    // 8) head -> out_x
    gcn_head<<<(N_NODES + 255) / 256, 256, 0, stream>>>(buf1, Wl1, bl1, Wl2, bl2, out_x);
}